// CliffordKinematicChain_62113817034794
// MI455X (gfx1250) — compile-verified
//
#include <hip/hip_runtime.h>
#include <stdint.h>

typedef float v2f __attribute__((ext_vector_type(2)));
typedef float v8f __attribute__((ext_vector_type(8)));

// ---------------- compile-time Clifford tables, sig = (+,+,+,+,-) ----------------
constexpr int SIGI[5] = {1, 1, 1, 1, -1};

constexpr int cay_sign(int a, int b) {
  // sign s with e_a * e_b = s * e_{a^b}
  int s = 1, res = a;
  for (int i = 0; i < 5; ++i) {
    if ((b >> i) & 1) {
      int c = 0;
      for (int j = i + 1; j < 5; ++j) c += (res >> j) & 1;
      if (c & 1) s = -s;
      if ((res >> i) & 1) { s *= SIGI[i]; res &= ~(1 << i); }
      else res |= (1 << i);
    }
  }
  return s;
}
constexpr int popc5(int x) { int c = 0; for (int i = 0; i < 5; ++i) c += (x >> i) & 1; return c; }

struct Tables {
  signed char sgn[32][32];   // Cayley sign
  signed char ns[32];        // NORM_SIGN = REV * SQ
  signed char c2s[3][32];    // stage-2 sign: s(q, q^K)*rev[q^K], K in {1,2,4}
  float t2[32][128];         // T2[c][j*32+q] = s(c, c^q) * templ[j][c^q]
};

constexpr Tables makeTables() {
  Tables T{};
  int rev[32] = {};
  for (int c = 0; c < 32; ++c) {
    int pc = popc5(c);
    rev[c] = ((pc * (pc - 1) / 2) % 2) ? -1 : 1;
  }
  for (int a = 0; a < 32; ++a)
    for (int b = 0; b < 32; ++b)
      T.sgn[a][b] = (signed char)cay_sign(a, b);
  for (int k = 0; k < 32; ++k)
    T.ns[k] = (signed char)(rev[k] * cay_sign(k, k));
  // conformal lift of backbone templates: blades 1,2,4 = x,y,z; 8 = .5r^2-.5; 16 = .5r^2+.5
  float tpl[4][32] = {};
  const float BB[4][3] = {{-0.529f, -1.359f, 1e-06f},
                          {0.f, 0.f, 0.f},
                          {1.525f, 0.f, 0.f},
                          {2.153f, 1.062f, 1e-06f}};
  for (int j = 0; j < 4; ++j) {
    float x = BB[j][0], y = BB[j][1], z = BB[j][2];
    tpl[j][1] = x; tpl[j][2] = y; tpl[j][4] = z;
    float r2 = x * x + y * y + z * z;
    tpl[j][8]  = 0.5f * r2 - 0.5f;
    tpl[j][16] = 0.5f * r2 + 0.5f;
  }
  for (int c = 0; c < 32; ++c)
    for (int j = 0; j < 4; ++j)
      for (int q = 0; q < 32; ++q)
        T.t2[c][j * 32 + q] = (float)cay_sign(c, c ^ q) * tpl[j][c ^ q];
  const int KSP[3] = {1, 2, 4};
  for (int k = 0; k < 3; ++k)
    for (int q = 0; q < 32; ++q)
      T.c2s[k][q] = (signed char)(cay_sign(q, q ^ KSP[k]) * rev[q ^ KSP[k]]);
  return T;
}
constexpr Tables gT = makeTables();

// ---------------- CDNA5 async global -> LDS helpers ----------------
__device__ __forceinline__ uint32_t lds_off(const void* p) {
  return (uint32_t)(uintptr_t)p;   // low 32 bits of flat LDS address = LDS byte offset
}
__device__ __forceinline__ void async_b128(uint32_t dst_lds, const float* g) {
  asm volatile("global_load_async_to_lds_b128 %0, %1, off"
               :: "v"(dst_lds), "v"((uint64_t)(uintptr_t)g)
               : "memory");
}
__device__ __forceinline__ void wait_async0() {
  asm volatile("s_wait_asynccnt 0" ::: "memory");
}

// ---------------- normalize with indefinite quadratic form ----------------
__device__ __forceinline__ void norm32(float v[32]) {
  float q = 0.f;
#pragma unroll
  for (int k = 0; k < 32; ++k)
    q = __builtin_fmaf(gT.ns[k] > 0 ? v[k] : -v[k], v[k], q);
  float rs = rsqrtf(fabsf(q) + 1e-12f);
#pragma unroll
  for (int k = 0; k < 32; ++k) v[k] *= rs;
}

// acc += gp(cur, part): acc[k] += s(k^b,b) * cur[k^b] * part[b]
// signs fold to fma neg-modifiers; backend packs pairs into v_pk_fma_f32
__device__ __forceinline__ void gp_acc(const float cur[32], const float part[32], float acc[32]) {
#pragma unroll
  for (int b = 0; b < 32; ++b) {
    const float yb = part[b];
#pragma unroll
    for (int k = 0; k < 32; ++k) {
      const float xa = cur[k ^ b];
      acc[k] = __builtin_fmaf(gT.sgn[k ^ b][b] > 0 ? xa : -xa, yb, acc[k]);
    }
  }
}

// ================= Kernel 1: per-batch Kogge-Stone scan of normalized geometric products
// block = one batch row: 256 threads, thread t owns chain[t]; val[t] <- val[t] (x) val[t-2^r]
__global__ void __launch_bounds__(256)
chain_scan_kernel(const float* __restrict__ motors, float* __restrict__ chain) {
  __shared__ float sm[256 * 33];          // padded rows: bank-conflict-free partner reads
  const int t = threadIdx.x;
  const float* src = motors + (size_t)blockIdx.x * 8192;

  // 32 KB tile: async copy global -> LDS (8 x b128 per thread), padded row layout
#pragma unroll
  for (int it = 0; it < 8; ++it) {
    int e = (it * 256 + t) * 4;           // float index, 16B aligned, within one 32f row
    int row = e >> 5, col = e & 31;
    async_b128(lds_off(&sm[row * 33 + col]), src + e);
  }
  wait_async0();
  __syncthreads();

  float cur[32];
#pragma unroll
  for (int w = 0; w < 32; ++w) cur[w] = sm[t * 33 + w];
  norm32(cur);                            // chain[0] = normalize(mot_0); scale-invariance makes
#pragma unroll                            // partner scaling irrelevant, only final must be normed
  for (int w = 0; w < 32; ++w) sm[t * 33 + w] = cur[w];
  __syncthreads();

  for (int r = 0; r < 8; ++r) {
    const int off = 1 << r;
    const bool act = (t >= off);
    float part[32];
    if (act) {
#pragma unroll
      for (int w = 0; w < 32; ++w) part[w] = sm[(t - off) * 33 + w];
    }
    __syncthreads();                      // all reads done before any write this round
    if (act) {
      float acc[32];
#pragma unroll
      for (int w = 0; w < 32; ++w) acc[w] = 0.f;
      gp_acc(cur, part, acc);             // later range on the LEFT: mot_s ... mot_0 ordering
      norm32(acc);
#pragma unroll
      for (int w = 0; w < 32; ++w) { cur[w] = acc[w]; sm[t * 33 + w] = acc[w]; }
    }
    __syncthreads();
  }

  // coalesced writeback of the normalized chain
  float* dst = chain + (size_t)blockIdx.x * 8192;
#pragma unroll 4
  for (int it = 0; it < 32; ++it) {
    int e = it * 256 + t;
    int row = e >> 5, col = e & 31;
    dst[e] = sm[row * 33 + col];
  }
}

// ================= Kernel 2: sandwich. U = M x T2 via V_WMMA_F32_16X16X4_F32 with
// T2 staged once per block into LDS (B-frags become ds_loads, no loadcnt stalls in the
// WMMA chain), then coords[j][k] = sum_q U[i,j*32+q] * c2s[k][q] * m[q^K] on VALU.
__global__ void __launch_bounds__(128)
sandwich_kernel(const float* __restrict__ chain, float* __restrict__ coords, int ntiles) {
  __shared__ float smT2[32 * 132];        // block-shared constant T2, padded rows
  __shared__ float smA[4][16 * 33 + 8];   // per-wave M tile (16 rows x 32, stride 33)
  __shared__ float smU[4][16 * 132];      // per-wave U tile (16 rows x 128, stride 132)
  const int tid  = threadIdx.x;
  const int lane = tid & 31;
  const int wid  = tid >> 5;

  // cooperative preload of T2 (4096 floats) into LDS once per block
#pragma unroll
  for (int it = 0; it < 32; ++it) {
    int idx = it * 128 + tid;             // 0..4095
    int r = idx >> 7, c = idx & 127;
    smT2[r * 132 + c] = gT.t2[r][c];
  }
  __syncthreads();                        // only block-wide barrier; waves independent after

  const int tile = blockIdx.x * 4 + wid;  // one 16-row tile per wave (wave-uniform guard)
  if (tile >= ntiles) return;
  float* A = smA[wid];
  float* U = smU[wid];
  const size_t i0 = (size_t)tile * 16;
  const float* g = chain + i0 * 32;

  // stage 2 KB tile async into this wave's LDS slab
#pragma unroll
  for (int it = 0; it < 4; ++it) {
    int e = (it * 32 + lane) * 4;         // 0..508
    int row = e >> 5, col = e & 31;
    async_b128(lds_off(&A[row * 33 + col]), g + e);
  }
  wait_async0();

  // A fragments (16x4 f32, doc layout: v0 = K+0 / K+2, v1 = K+1 / K+3 by lane half)
  const int arow = lane & 15, ah = lane >> 4;
  v2f afrag[8];
#pragma unroll
  for (int kk = 0; kk < 8; ++kk) {
    int kb = kk * 4 + ah * 2;
    afrag[kk].x = A[arow * 33 + kb];
    afrag[kk].y = A[arow * 33 + kb + 1];
  }

  const int crow0 = ah * 8, ccol = lane & 15;
#pragma unroll
  for (int nt = 0; nt < 8; ++nt) {        // 8 N-tiles of 16 cols (j,q)
    // B fragments for this N-tile from LDS-resident T2
    v2f bfrag[8];
#pragma unroll
    for (int kk = 0; kk < 8; ++kk) {
      int kb = kk * 4 + ah * 2;
      bfrag[kk].x = smT2[kb * 132 + nt * 16 + ccol];
      bfrag[kk].y = smT2[(kb + 1) * 132 + nt * 16 + ccol];
    }
    v8f acc = {0.f, 0.f, 0.f, 0.f, 0.f, 0.f, 0.f, 0.f};
#pragma unroll
    for (int kk = 0; kk < 8; ++kk) {      // K = 32 in steps of 4
      acc = __builtin_amdgcn_wmma_f32_16x16x4_f32(
          false, afrag[kk], false, bfrag[kk], (short)0, acc, false, false);
    }
    // scatter C fragment to U tile per doc layout (VGPR r, lane half -> row)
#pragma unroll
    for (int rr = 0; rr < 8; ++rr)
      U[(crow0 + rr) * 132 + nt * 16 + ccol] = acc[rr];
  }

  // stage 2: lane pair per row; each lane does 2 templates x 3 spatial blades
  const int iloc = lane >> 1;
  const int jp = (lane & 1) * 2;
  float m[32];
#pragma unroll
  for (int w = 0; w < 32; ++w) m[w] = A[iloc * 33 + w];

  float res[6];
#pragma unroll
  for (int jj = 0; jj < 2; ++jj) {
    const int j = jp + jj;
    float p0 = 0.f, p1 = 0.f, p2 = 0.f;
#pragma unroll
    for (int q = 0; q < 32; ++q) {
      const float u = U[iloc * 132 + j * 32 + q];
      p0 = __builtin_fmaf(gT.c2s[0][q] > 0 ? u : -u, m[q ^ 1], p0);
      p1 = __builtin_fmaf(gT.c2s[1][q] > 0 ? u : -u, m[q ^ 2], p1);
      p2 = __builtin_fmaf(gT.c2s[2][q] > 0 ? u : -u, m[q ^ 4], p2);
    }
    res[jj * 3 + 0] = p0; res[jj * 3 + 1] = p1; res[jj * 3 + 2] = p2;
  }
  float* dst = coords + (i0 + iloc) * 12 + jp * 3;
#pragma unroll
  for (int w = 0; w < 6; ++w) dst[w] = res[w];
}

// ================= host launch =================
extern "C" void kernel_launch(void* const* d_in, const int* in_sizes, int n_in,
                              void* d_out, int out_size, void* d_ws, size_t ws_size,
                              hipStream_t stream) {
  (void)n_in; (void)out_size; (void)d_ws; (void)ws_size;
  const float* motors = (const float*)d_in[0];
  float* out = (float*)d_out;

  const long n    = (long)in_sizes[0];  // B*256*32
  const long rows = n / 32;             // B*256 multivectors
  const long B    = rows / 256;

  float* coords = out;                  // (B,256,4,3)
  float* chain  = out + rows * 12;      // (B,256,32)

  chain_scan_kernel<<<dim3((unsigned)B), dim3(256), 0, stream>>>(motors, chain);

  const int ntiles = (int)(rows / 16);
  const int blocks = (ntiles + 3) / 4;
  sandwich_kernel<<<dim3(blocks), dim3(128), 0, stream>>>(chain, coords, ntiles);
}